// CTCLoss_37400575214061
// MI455X (gfx1250) — compile-verified
//
#include <hip/hip_runtime.h>
#include <stdint.h>

// Problem dims fixed by the reference setup_inputs():
#define T_DIM 1000
#define B_DIM 32
#define C_DIM 2000
#define L_DIM 100
#define S_DIM 201            // 2*L + 1
#define NEGV  (-1e30f)

typedef uint32_t u32;
typedef uint64_t u64;
typedef u32 v4u  __attribute__((ext_vector_type(4)));
typedef int  v4i  __attribute__((ext_vector_type(4)));
typedef int  v8i  __attribute__((ext_vector_type(8)));

// ---------------------------------------------------------------------------
// TDM: DMA a contiguous run of `nelems` f32 from global memory into LDS.
// Descriptor layout per CDNA5 ISA §8 (D# groups 0 & 1), 1-D tile:
//   data_size=4B, tensor_dim0 = tile_dim0 = nelems, dim1 = 1.
// 6-arg builtin (clang-23 / therock-10.0 form); lowers to tensor_load_to_lds.
// ---------------------------------------------------------------------------
__device__ __forceinline__ void tdm_load_1d(const float* gsrc, u32 lds_byte_off,
                                            u32 nelems) {
  u64 ga = (u64)(uintptr_t)gsrc;
  v4u g0;
  g0[0] = 1u;                                   // count=1 (valid), user mode
  g0[1] = lds_byte_off;                         // LDS byte address
  g0[2] = (u32)(ga & 0xffffffffu);              // global_addr[31:0]
  g0[3] = (u32)((ga >> 32) & 0x1ffffffu)        // global_addr[56:32]
        | (2u << 30);                           // type = 2 ("image")
  u32 dim0 = nelems, dim1 = 1;
  u32 tile0 = nelems, tile1 = 1, tile2 = 0;     // tile0 must fit 16 bits
  u64 str0 = nelems, str1 = 0;
  v8i g1;
  g1[0] = (int)(2u << 16);                      // workgroup_mask=0, data_size=2 (4B)
  g1[1] = (int)((dim0 & 0xffffu) << 16);        // tensor_dim0[15:0] @ bit48
  g1[2] = (int)((dim0 >> 16) | ((dim1 & 0xffffu) << 16));
  g1[3] = (int)((dim1 >> 16) | (tile0 << 16));  // tile_dim0 @ bit112
  g1[4] = (int)(tile1 | (tile2 << 16));
  g1[5] = (int)(str0 & 0xffffffffu);            // tensor_dim0_stride[31:0]
  g1[6] = (int)(((str0 >> 32) & 0xffffu) | ((str1 & 0xffffu) << 16));
  g1[7] = (int)((str1 >> 16) & 0xffffffffu);
  v4i g2 = {0, 0, 0, 0};                        // unused (<=2D tensor)
  v4i g3 = {0, 0, 0, 0};
  v8i g4 = {0, 0, 0, 0, 0, 0, 0, 0};            // extra group (6-arg form)
  __builtin_amdgcn_tensor_load_to_lds(g0, g1, g2, g3, g4, 0);
}

// ---------------------------------------------------------------------------
// Kernel 1: one wave per (t,b) row of activations.
//   TDM row -> LDS; two-pass logsumexp from LDS (float4 / ds_load_b128);
//   then gather the 201 extended-label log-probs from LDS and write
//   lp_ext[(t*B+b)*S + s]. 8 waves/block, 8 rows/block, 64000 B static LDS.
//   No block barriers needed: each wave owns its LDS slice end-to-end.
// ---------------------------------------------------------------------------
__global__ __launch_bounds__(256) void ctc_row_kernel(
    const float* __restrict__ acts,       // [T*B*C]
    const int*   __restrict__ targets,    // [B*L]
    float*       __restrict__ lp_ext) {   // [T*B*S]
  __shared__ __align__(16) float smem[8][C_DIM];
  const int lane = (int)(threadIdx.x & 31u);
  const int wv   = __builtin_amdgcn_readfirstlane((int)(threadIdx.x >> 5));
  const int row  = (int)blockIdx.x * 8 + wv;          // row = t*B + b
  const float* src = acts + (size_t)row * C_DIM;

  // Async DMA this wave's row into its LDS slice, then wait on TENSORcnt.
  tdm_load_1d(src, (u32)(uintptr_t)(&smem[wv][0]), C_DIM);
  __builtin_amdgcn_s_wait_tensorcnt(0);

  const float4* rowv = reinterpret_cast<const float4*>(&smem[wv][0]);

  // Pass 1: row max (vectorized LDS reads + wave32 shuffle tree).
  float m = NEGV;
  for (int c = lane; c < C_DIM / 4; c += 32) {
    const float4 v = rowv[c];
    m = fmaxf(m, fmaxf(fmaxf(v.x, v.y), fmaxf(v.z, v.w)));
  }
  #pragma unroll
  for (int off = 16; off > 0; off >>= 1) m = fmaxf(m, __shfl_xor(m, off, 32));

  // Pass 2: sum of exp(x - m).
  float ssum = 0.0f;
  for (int c = lane; c < C_DIM / 4; c += 32) {
    const float4 v = rowv[c];
    ssum += expf(v.x - m) + expf(v.y - m) + expf(v.z - m) + expf(v.w - m);
  }
  #pragma unroll
  for (int off = 16; off > 0; off >>= 1) ssum += __shfl_xor(ssum, off, 32);
  const float lse = m + logf(ssum);

  // Gather extended-label log-probs straight out of LDS (row still resident).
  const int b = row & (B_DIM - 1);                    // B = 32
  const int* tgt = targets + b * L_DIM;
  float* out = lp_ext + (size_t)row * S_DIM;
  for (int s = lane; s < S_DIM; s += 32) {
    const int e = (s & 1) ? tgt[s >> 1] : 0;          // blank = 0
    out[s] = smem[wv][e] - lse;
  }
}

// ---------------------------------------------------------------------------
// Kernel 2: alpha DP. One block per utterance, thread s owns lattice state s.
// Ping-pong alpha rows in LDS; one barrier per time step; prefetch next row.
// ---------------------------------------------------------------------------
__global__ __launch_bounds__(256) void ctc_alpha_kernel(
    const float* __restrict__ lp_ext,         // [T*B*S]
    const int*   __restrict__ targets,        // [B*L]
    const int*   __restrict__ input_lengths,  // [B]
    const int*   __restrict__ target_lengths, // [B]
    float*       __restrict__ costs) {        // [B]
  const int b = (int)blockIdx.x;
  const int s = (int)threadIdx.x;
  __shared__ float abuf[2][S_DIM + 1];
  __shared__ unsigned char skipm[S_DIM];

  const int* tgt = targets + b * L_DIM;
  if (s < S_DIM) {
    bool sk = false;
    if (s & 1) {                                   // ext[s] = tgt[s>>1] != blank
      const int e = tgt[s >> 1];
      sk = (s == 1) || (e != tgt[(s >> 1) - 1]);   // ext[s] != ext[s-2]
    }
    skipm[s] = (unsigned char)sk;
    abuf[0][s] = (s < 2) ? lp_ext[(size_t)b * S_DIM + s] : NEGV;  // t = 0
  }
  __syncthreads();

  const int Tb = input_lengths[b];
  int p = 0;
  for (int t = 1; t < Tb; ++t) {
    const float* lpr = lp_ext + ((size_t)t * B_DIM + b) * S_DIM;
    float nv = NEGV;
    if (s < S_DIM) {
      const float lp = lpr[s];
      if (t + 1 < Tb)                              // hide next row's latency
        __builtin_prefetch(lp_ext + ((size_t)(t + 1) * B_DIM + b) * S_DIM + s, 0, 3);
      const float a  = abuf[p][s];
      const float a1 = (s >= 1) ? abuf[p][s - 1] : NEGV;
      const float a2 = (s >= 2 && skipm[s]) ? abuf[p][s - 2] : NEGV;
      const float mm = fmaxf(a, fmaxf(a1, a2));
      nv = mm + logf(expf(a - mm) + expf(a1 - mm) + expf(a2 - mm)) + lp;
    }
    if (s < S_DIM) abuf[1 - p][s] = nv;            // write other buffer: no race
    __syncthreads();
    p ^= 1;
  }

  if (s == 0) {
    const int sl = 2 * target_lengths[b];          // final blank position
    const float ab = abuf[p][sl];
    const float al = abuf[p][sl - 1];
    const float mm = fmaxf(ab, al);
    costs[b] = -(mm + logf(expf(ab - mm) + expf(al - mm)));
  }
}

// ---------------------------------------------------------------------------
// Kernel 3: warp-ctc 'mean' reduction: sum(costs) / B / sum(target_lengths).
// ---------------------------------------------------------------------------
__global__ void ctc_reduce_kernel(const float* __restrict__ costs,
                                  const int* __restrict__ target_lengths,
                                  float* __restrict__ out) {
  const int lane = (int)threadIdx.x;               // exactly one wave32, B==32
  float c = costs[lane];
  int  tl = target_lengths[lane];
  #pragma unroll
  for (int off = 16; off > 0; off >>= 1) {
    c  += __shfl_xor(c, off, 32);
    tl += __shfl_xor(tl, off, 32);
  }
  if (lane == 0) out[0] = c / (float)B_DIM / (float)tl;
}

// ---------------------------------------------------------------------------
// Workspace layout (requires ws_size >= T*B*S*4 + B*4 ≈ 24.6 MiB):
//   [0, T*B*S) floats : lp_ext
//   then B floats     : per-utterance costs
// ---------------------------------------------------------------------------
extern "C" void kernel_launch(void* const* d_in, const int* in_sizes, int n_in,
                              void* d_out, int out_size, void* d_ws, size_t ws_size,
                              hipStream_t stream) {
  (void)in_sizes; (void)n_in; (void)out_size; (void)ws_size;
  const float* acts    = (const float*)d_in[0];
  const int*   targets = (const int*)d_in[1];
  const int*   in_len  = (const int*)d_in[2];
  const int*   tg_len  = (const int*)d_in[3];
  float* outp   = (float*)d_out;
  float* lp_ext = (float*)d_ws;
  float* costs  = lp_ext + (size_t)T_DIM * B_DIM * S_DIM;

  ctc_row_kernel  <<<(T_DIM * B_DIM) / 8, 256, 0, stream>>>(acts, targets, lp_ext);
  ctc_alpha_kernel<<<B_DIM,             256, 0, stream>>>(lp_ext, targets, in_len,
                                                          tg_len, costs);
  ctc_reduce_kernel<<<1, 32, 0, stream>>>(costs, tg_len, outp);
}